// SAGE_57294863728943
// MI455X (gfx1250) — compile-verified
//
#include <hip/hip_runtime.h>
#include <math.h>

// ---------------------------------------------------------------------------
// GraphSAGE (2x SAGEConv + Linear), d=64 everywhere, f32.
// Aggregation: L2-resident gather + atomic scatter-add (graph fits in 192MB L2).
// Dense parts: V_WMMA_F32_16X16X4_F32, one 16-row strip per wave32.
// ---------------------------------------------------------------------------

typedef float v2f __attribute__((ext_vector_type(2)));
typedef float v8f __attribute__((ext_vector_type(8)));

#define NFEAT 64

// ---------------------------------------------------------------------------
__global__ void sage_zero_f32(float* __restrict__ p, long n) {
  long i = (long)blockIdx.x * blockDim.x + threadIdx.x;
  long stride = (long)gridDim.x * blockDim.x;
  for (; i < n; i += stride) p[i] = 0.0f;
}

// 16 threads per edge; each thread handles 4 consecutive features (float4).
__global__ void sage_scatter(const float* __restrict__ feat,
                             const int* __restrict__ src,
                             const int* __restrict__ dst,
                             float* __restrict__ nbr_sum,
                             float* __restrict__ cnt,
                             int E, int add_cnt) {
  long tid = (long)blockIdx.x * blockDim.x + threadIdx.x;
  int e = (int)(tid >> 4);
  if (e >= E) return;
  int g = (int)(tid & 15);

  int s = src[e];
  int d = dst[e];
  const float4 v = *(const float4*)(feat + (long)s * NFEAT + g * 4);
  float* o = nbr_sum + (long)d * NFEAT + g * 4;
  atomicAdd(o + 0, v.x);
  atomicAdd(o + 1, v.y);
  atomicAdd(o + 2, v.z);
  atomicAdd(o + 3, v.w);
  if (add_cnt && g == 0) atomicAdd(cnt + d, 1.0f);
}

// ---------------------------------------------------------------------------
// out[strip..strip+15, 0:64] = act( mean(nbr) @ Wl + bias + xin @ Wr )
// One wave per 16-row strip. Accumulators: 4 tiles of 16x16 f32 (v8f each).
//
// Fragment layouts (per CDNA5 ISA 7.12.2, f32, wave32):
//   A (16x4, MxK):  lane l holds A[l%16][2*(l/16) + {0,1}]        (2 VGPRs)
//   B (4x16, KxN):  lane l holds B[2*(l/16) + {0,1}][l%16]        (2 VGPRs)
//   C/D (16x16)  :  lane l, vgpr v holds D[v + 8*(l/16)][l%16]    (8 VGPRs)
//
// USE_NBR / ELU are template params: straight-line inner loop, no per-k
// branches, branchless ELU epilogue.
// ---------------------------------------------------------------------------
template <bool USE_NBR, bool ELU>
__global__ __launch_bounds__(256) void sage_gemm(
    const float* xin,                     // [N,64] root features (may alias out)
    const float* __restrict__ nbr_sum,    // [N,64] (USE_NBR only)
    const float* __restrict__ cnt,        // [N]    (USE_NBR only)
    const float* __restrict__ Wl,         // [64,64] neighbor weight (K x N)
    const float* __restrict__ Wr,         // [64,64] root weight
    const float* __restrict__ bias,       // [64]
    float* out,                           // [N,64] (may alias xin)
    int nrows) {
  __shared__ float sWl[NFEAT * NFEAT];
  __shared__ float sWr[NFEAT * NFEAT];
  __shared__ float sB[NFEAT];

  for (int i = threadIdx.x; i < NFEAT * NFEAT; i += 256) {
    sWr[i] = Wr[i];
    if (USE_NBR) sWl[i] = Wl[i];
  }
  if (threadIdx.x < NFEAT) sB[threadIdx.x] = bias[threadIdx.x];
  __syncthreads();

  const int wave = threadIdx.x >> 5;
  const int lane = threadIdx.x & 31;
  const int n    = lane & 15;   // column within tile / row index m for A-frag
  const int half = lane >> 4;   // selects K-pair

  const int strip = (blockIdx.x * 8 + wave) * 16;
  if (strip >= nrows) return;   // wave-uniform: EXEC stays all-ones for WMMA

  const float* xrow = xin + (long)(strip + n) * NFEAT;
  const float* srow = USE_NBR ? nbr_sum + (long)(strip + n) * NFEAT : nullptr;
  float rdeg = 0.0f;
  if (USE_NBR) rdeg = 1.0f / fmaxf(cnt[strip + n], 1.0f);

  // Init accumulators with bias (depends only on output column).
  v8f acc[4];
#pragma unroll
  for (int t = 0; t < 4; ++t) {
    float bv = sB[t * 16 + n];
#pragma unroll
    for (int v = 0; v < 8; ++v) acc[t][v] = bv;
  }

#pragma unroll
  for (int kk = 0; kk < 16; ++kk) {
    const int k0 = kk * 4 + half * 2;     // this lane's K-pair base
    v2f ar = *(const v2f*)(xrow + k0);    // root A-fragment
    v2f an;
    if (USE_NBR) {
      v2f s = *(const v2f*)(srow + k0);   // neighbor-mean A-fragment
      an.x = s.x * rdeg;
      an.y = s.y * rdeg;
    }
#pragma unroll
    for (int t = 0; t < 4; ++t) {
      v2f br;
      br.x = sWr[(k0 + 0) * NFEAT + t * 16 + n];
      br.y = sWr[(k0 + 1) * NFEAT + t * 16 + n];
      acc[t] = __builtin_amdgcn_wmma_f32_16x16x4_f32(
          false, ar, false, br, (short)0, acc[t], false, false);
      if (USE_NBR) {
        v2f bl;
        bl.x = sWl[(k0 + 0) * NFEAT + t * 16 + n];
        bl.y = sWl[(k0 + 1) * NFEAT + t * 16 + n];
        acc[t] = __builtin_amdgcn_wmma_f32_16x16x4_f32(
            false, an, false, bl, (short)0, acc[t], false, false);
      }
    }
  }

  // Epilogue: branchless ELU, store D tiles. All xin reads above complete
  // before any store, so in-place (final Linear) is safe per-strip.
#pragma unroll
  for (int t = 0; t < 4; ++t) {
#pragma unroll
    for (int v = 0; v < 8; ++v) {
      float val = acc[t][v];
      if (ELU) {
        // elu(x) = max(x,0) + exp(min(x,0)) - 1   (exact on both sides)
        val = fmaxf(val, 0.0f) + __expf(fminf(val, 0.0f)) - 1.0f;
      }
      const int row = strip + v + half * 8;
      out[(long)row * NFEAT + t * 16 + n] = val;
    }
  }
}

// ---------------------------------------------------------------------------
extern "C" void kernel_launch(void* const* d_in, const int* in_sizes, int n_in,
                              void* d_out, int out_size, void* d_ws, size_t ws_size,
                              hipStream_t stream) {
  const float* x    = (const float*)d_in[0];
  const int*   ei   = (const int*)d_in[1];
  const float* W1l  = (const float*)d_in[2];
  const float* b1   = (const float*)d_in[3];
  const float* W1r  = (const float*)d_in[4];
  const float* W2l  = (const float*)d_in[5];
  const float* b2   = (const float*)d_in[6];
  const float* W2r  = (const float*)d_in[7];
  const float* Wlin = (const float*)d_in[8];
  const float* blin = (const float*)d_in[9];

  const int N = in_sizes[0] / NFEAT;   // 100000
  const int E = in_sizes[1] / 2;       // 1600000
  const int* src = ei;
  const int* dst = ei + E;

  // Workspace layout: [nbr_sum N*64][cnt N][h1 N*64]
  float* nbr_sum = (float*)d_ws;
  float* cnt     = nbr_sum + (size_t)N * NFEAT;
  float* h1      = cnt + N;
  float* hout    = (float*)d_out;

  const int scat_blocks = (int)(((long)E * 16 + 255) / 256);
  const int strips      = (N + 15) / 16;
  const int gemm_blocks = (strips + 7) / 8;

  // Layer 1: zero sums+counts (contiguous), scatter, fused GEMM+ELU -> h1
  sage_zero_f32<<<1024, 256, 0, stream>>>(nbr_sum, (long)N * NFEAT + N);
  sage_scatter<<<scat_blocks, 256, 0, stream>>>(x, src, dst, nbr_sum, cnt, E, 1);
  sage_gemm<true, true><<<gemm_blocks, 256, 0, stream>>>(
      x, nbr_sum, cnt, W1l, W1r, b1, h1, N);

  // Layer 2: re-zero sums only (degrees unchanged), scatter h1, GEMM -> d_out
  sage_zero_f32<<<1024, 256, 0, stream>>>(nbr_sum, (long)N * NFEAT);
  sage_scatter<<<scat_blocks, 256, 0, stream>>>(h1, src, dst, nbr_sum, cnt, E, 0);
  sage_gemm<true, true><<<gemm_blocks, 256, 0, stream>>>(
      h1, nbr_sum, cnt, W2l, W2r, b2, hout, N);

  // Final Linear, in place on d_out (each wave reads its strip before storing).
  sage_gemm<false, false><<<gemm_blocks, 256, 0, stream>>>(
      hout, nullptr, nullptr, nullptr, Wlin, blin, hout, N);
}